// SparseMHA_41755672052281
// MI455X (gfx1250) — compile-verified
//
#include <hip/hip_runtime.h>
#include <math.h>

// Problem constants (match reference)
#define N_NODES 50000
#define DEG     16
#define H       128
#define NH      8
#define HD      16
#define SCALE_F 0.25f          // HD^-0.5
#define MTILES  (N_NODES / 16) // 3125 exactly

typedef __attribute__((ext_vector_type(2))) float v2f;
typedef __attribute__((ext_vector_type(8))) float v8f;

#define LDA 132   // padded LDS row (floats) for the 16x128 A tile (bank-conflict-free b64 reads)
#define LDK 136   // padded LDS row (floats) for gathered K/V rows

// ---------------------------------------------------------------------------
// Kernel A: fused Q/K/V projection  P = h @ W.T + b  via V_WMMA_F32_16X16X4_F32
// Persistent blocks: 8 waves/block, wave `w` owns column tile ct = w (16 cols),
// holds all B fragments (W.T) for Wq/Wk/Wv in registers, loops over M tiles.
// ---------------------------------------------------------------------------
__global__ __launch_bounds__(256) void qkv_wmma_kernel(
    const float* __restrict__ hsrc,
    const float* __restrict__ Wq, const float* __restrict__ bq,
    const float* __restrict__ Wk, const float* __restrict__ bk,
    const float* __restrict__ Wv, const float* __restrict__ bv,
    float* __restrict__ Qp, float* __restrict__ Kp, float* __restrict__ Vp)
{
    __shared__ float sA[16 * LDA];

    const int t    = threadIdx.x;
    const int ct   = t >> 5;       // wave id = column tile (0..7)
    const int lane = t & 31;
    const int nloc = lane & 15;    // N (cols) / M-row-in-tile position
    const int hi   = lane >> 4;    // 0: K={0,1} half, 1: K={2,3} half
    const int colB = ct * 16 + nloc;
    const int kOff = hi << 1;

    // Preload B fragments for all 32 K-steps, all three weight matrices.
    // B[k, n] = W[n, k]  (since P = h @ W.T). Layout (4x16 f32 B, wave32):
    //   VGPR0: lanes0-15 row K=k0, lanes16-31 row K=k0+2
    //   VGPR1: lanes0-15 row K=k0+1, lanes16-31 row K=k0+3
    // -> lane holds W[colB, k0+kOff], W[colB, k0+kOff+1] as a v2f.
    v2f Bq[32], Bk[32], Bv[32];
#pragma unroll
    for (int kt = 0; kt < 32; ++kt) {
        const int k0 = kt * 4 + kOff;
        Bq[kt] = *(const v2f*)(Wq + (size_t)colB * H + k0);
        Bk[kt] = *(const v2f*)(Wk + (size_t)colB * H + k0);
        Bv[kt] = *(const v2f*)(Wv + (size_t)colB * H + k0);
    }
    const float bqv = bq[colB];
    const float bkv = bk[colB];
    const float bvv = bv[colB];

    for (int m = blockIdx.x; m < MTILES; m += gridDim.x) {
        __syncthreads();  // protect sA reuse across M iterations
        // Stage 16x128 tile of h into LDS (coalesced: 2 rows per pass).
        for (int i = t; i < 16 * H; i += 256) {
            const int r = i >> 7, c = i & 127;
            sA[r * LDA + c] = hsrc[(size_t)(m * 16 + r) * H + c];
        }
        __syncthreads();

        v8f aq = {}, ak = {}, av = {};
#if __has_builtin(__builtin_amdgcn_wmma_f32_16x16x4_f32)
#pragma unroll
        for (int kt = 0; kt < 32; ++kt) {
            // A fragment (16x4 f32): lanes0-15 rows M=0..15 K={k0,k0+1},
            //                        lanes16-31 rows M=0..15 K={k0+2,k0+3}
            v2f a = *(const v2f*)(&sA[nloc * LDA + kt * 4 + kOff]);
            aq = __builtin_amdgcn_wmma_f32_16x16x4_f32(false, a, false, Bq[kt],
                                                       (short)0, aq, false, false);
            ak = __builtin_amdgcn_wmma_f32_16x16x4_f32(false, a, false, Bk[kt],
                                                       (short)0, ak, false, false);
            av = __builtin_amdgcn_wmma_f32_16x16x4_f32(false, a, false, Bv[kt],
                                                       (short)0, av, false, false);
        }
#else
        // VALU fallback (same D-fragment register layout as WMMA path)
#pragma unroll
        for (int v = 0; v < 8; ++v) {
            float s0 = 0.f, s1 = 0.f, s2 = 0.f;
            const int row = (hi << 3) + v;
            for (int k = 0; k < H; ++k) {
                const float a = sA[row * LDA + k];
                s0 += a * Wq[(size_t)colB * H + k];
                s1 += a * Wk[(size_t)colB * H + k];
                s2 += a * Wv[(size_t)colB * H + k];
            }
            aq[v] = s0; ak[v] = s1; av[v] = s2;
        }
#endif
        // D fragment 16x16 f32: lane = N + 16*(M>=8), VGPR v holds M = v (+8).
        const int mBase = m * 16 + (hi << 3);
#pragma unroll
        for (int v = 0; v < 8; ++v) {
            const size_t o = (size_t)(mBase + v) * H + colB;
            Qp[o] = aq[v] + bqv;
            Kp[o] = ak[v] + bkv;
            Vp[o] = av[v] + bvv;
        }
    }
}

// ---------------------------------------------------------------------------
// Kernel B: per-node edge phase. One 128-thread block per node.
// Head hh owns channels {c : c % 8 == hh} (from the (N, HD, NH) reshape).
// ---------------------------------------------------------------------------
__global__ __launch_bounds__(128) void edge_attn_kernel(
    const int* __restrict__ row_ptr, const int* __restrict__ col_ind,
    const float* __restrict__ Qp, const float* __restrict__ Kp,
    const float* __restrict__ Vp, float* __restrict__ out)
{
    __shared__ float sK[DEG * LDK];
    __shared__ float sV[DEG * LDK];
    __shared__ float sQ[H];
    __shared__ float sS[DEG * NH];
    __shared__ int   sCol[DEG];

    const int node = blockIdx.x;
    const int t    = threadIdx.x;
    const int base = row_ptr[node];

    if (t < DEG) sCol[t] = col_ind[base + t];
    sQ[t] = Qp[(size_t)node * H + t];
    __syncthreads();

    // Gather neighbor K/V rows with fully-coalesced 512B row loads (L2-hot).
    for (int j = 0; j < DEG; ++j) {
        const int c = sCol[j];
        sK[j * LDK + t] = Kp[(size_t)c * H + t];
        sV[j * LDK + t] = Vp[(size_t)c * H + t];
    }
    __syncthreads();

    // thread (nbr, head) -> one score = q[node,:,head] . k[nbr,:,head]
    const int nbr  = t >> 3;
    const int head = t & 7;
    float sc = 0.f;
#pragma unroll
    for (int d = 0; d < HD; ++d)
        sc += sQ[d * NH + head] * sK[nbr * LDK + d * NH + head];
    sc *= SCALE_F;

    sS[t] = sc;
    __syncthreads();
    float mx = -INFINITY;
#pragma unroll
    for (int j = 0; j < DEG; ++j) mx = fmaxf(mx, sS[j * NH + head]);
    __syncthreads();

    const float p = __expf(sc - mx);
    sS[t] = p;
    __syncthreads();
    float sum = 0.f;
#pragma unroll
    for (int j = 0; j < DEG; ++j) sum += sS[j * NH + head];
    __syncthreads();

    sS[t] = p / sum;   // attn[nbr, head]
    __syncthreads();

    // Remap: thread t == output channel c; attn index uses c % 8 == head.
    float acc = 0.f;
#pragma unroll
    for (int j = 0; j < DEG; ++j)
        acc += sS[j * NH + head] * sV[j * LDK + t];
    out[(size_t)node * H + t] = acc;
}

// ---------------------------------------------------------------------------
extern "C" void kernel_launch(void* const* d_in, const int* in_sizes, int n_in,
                              void* d_out, int out_size, void* d_ws, size_t ws_size,
                              hipStream_t stream)
{
    const float* h       = (const float*)d_in[0];
    const int*   row_ptr = (const int*)  d_in[1];
    const int*   col_ind = (const int*)  d_in[2];
    const float* Wq      = (const float*)d_in[3];
    const float* bq      = (const float*)d_in[4];
    const float* Wk      = (const float*)d_in[5];
    const float* bk      = (const float*)d_in[6];
    const float* Wv      = (const float*)d_in[7];
    const float* bv      = (const float*)d_in[8];
    float* out = (float*)d_out;

    float* Qp = (float*)d_ws;                       // N*H floats
    float* Kp = Qp + (size_t)N_NODES * H;           // N*H floats
    float* Vp = Kp + (size_t)N_NODES * H;           // N*H floats (total 76.8 MB)

    qkv_wmma_kernel<<<512, 256, 0, stream>>>(h, Wq, bq, Wk, bk, Wv, bv, Qp, Kp, Vp);
    edge_attn_kernel<<<N_NODES, 128, 0, stream>>>(row_ptr, col_ind, Qp, Kp, Vp, out);
}